// EarlyLearningRegularizationLoss_57062935495532
// MI455X (gfx1250) — compile-verified
//
#include <hip/hip_runtime.h>
#include <hip/hip_bf16.h>
#include <stdint.h>

// Early-Learning-Regularization loss, CDNA5 (gfx1250, wave32).
//
// Per row: m = max(l); e_i = exp(l_i - m); S = sum e; T1 = sum e*hist;
// T2 = sum e*e.  dot = seen ? 0.9*T1/S + 0.1*T2/S^2 : T2/S^2.
// partial[row] = (m + log S - l[target]) + 3*log(1 - dot + 1e-4).
// out = mean(partial).
//
// Data movement uses the CDNA5 async global->LDS path (ASYNCcnt).

#define LAMBDA_ELR 3.0f
#define EMA_OLD    0.9f
#define EMA_NEW    0.1f
#define EPS_ELR    1e-4f

#define BLOCK_THREADS 256
#define NWAVES (BLOCK_THREADS / 32)
#define MAX_C 1024   // C = 1000 in this problem

__device__ __forceinline__ float wave_sum(float v) {
#pragma unroll
  for (int m = 16; m > 0; m >>= 1) v += __shfl_xor(v, m, 32);
  return v;
}
__device__ __forceinline__ float wave_max(float v) {
#pragma unroll
  for (int m = 16; m > 0; m >>= 1) v = fmaxf(v, __shfl_xor(v, m, 32));
  return v;
}

__global__ __launch_bounds__(BLOCK_THREADS)
void elr_row_kernel(const float* __restrict__ logits,
                    const float* __restrict__ history,
                    const unsigned char* __restrict__ seen,
                    const int* __restrict__ targets,
                    const int* __restrict__ ids,
                    float* __restrict__ partial,
                    int C)
{
  __shared__ float s_logits[MAX_C];
  __shared__ float s_hist[MAX_C];
  __shared__ float s_m[NWAVES];
  __shared__ float s_s[NWAVES];
  __shared__ float s_t1[NWAVES];
  __shared__ float s_t2[NWAVES];
  __shared__ float s_bcast;

  const int row = blockIdx.x;
  const int tid = threadIdx.x;
  const int id  = ids[row];

  const float* lrow = logits  + (size_t)row * (size_t)C;
  const float* hrow = history + (size_t)id  * (size_t)C;

  // ---- stream both rows into LDS via CDNA5 async loads (ASYNCcnt) ----
  const int chunks = (C * 4) >> 4;  // number of 16-byte transfers per row
  const uint32_t l_lds = (uint32_t)(uintptr_t)(void*)s_logits;
  const uint32_t h_lds = (uint32_t)(uintptr_t)(void*)s_hist;
  for (int c = tid; c < chunks; c += BLOCK_THREADS) {
    const uint32_t off = (uint32_t)c << 4;
    asm volatile("global_load_async_to_lds_b128 %0, %1, %2"
                 :: "v"(l_lds + off), "v"(off), "s"(lrow) : "memory");
    asm volatile("global_load_async_to_lds_b128 %0, %1, %2"
                 :: "v"(h_lds + off), "v"(off), "s"(hrow) : "memory");
  }
  // tail (only if C % 4 != 0; C=1000 has no tail)
  for (int i = (chunks << 2) + tid; i < C; i += BLOCK_THREADS) {
    s_logits[i] = lrow[i];
    s_hist[i]   = hrow[i];
  }
  asm volatile("s_wait_asynccnt 0" ::: "memory");
  __syncthreads();

  // ---- pass A: row max ----
  float m = -3.402823466e+38f;
  for (int i = tid; i < C; i += BLOCK_THREADS) m = fmaxf(m, s_logits[i]);
  m = wave_max(m);
  const int wv = tid >> 5, ln = tid & 31;
  if (ln == 0) s_m[wv] = m;
  __syncthreads();
  if (tid == 0) {
    float mm = s_m[0];
#pragma unroll
    for (int w = 1; w < NWAVES; ++w) mm = fmaxf(mm, s_m[w]);
    s_bcast = mm;
  }
  __syncthreads();
  m = s_bcast;

  // ---- pass B: S, T1, T2 ----
  float s = 0.f, t1 = 0.f, t2 = 0.f;
  for (int i = tid; i < C; i += BLOCK_THREADS) {
    const float e = __expf(s_logits[i] - m);
    s += e;
    t2 = fmaf(e, e, t2);
    t1 = fmaf(e, s_hist[i], t1);
  }
  s  = wave_sum(s);
  t1 = wave_sum(t1);
  t2 = wave_sum(t2);
  if (ln == 0) { s_s[wv] = s; s_t1[wv] = t1; s_t2[wv] = t2; }
  __syncthreads();

  if (tid == 0) {
    float S = 0.f, T1 = 0.f, T2 = 0.f;
#pragma unroll
    for (int w = 0; w < NWAVES; ++w) { S += s_s[w]; T1 += s_t1[w]; T2 += s_t2[w]; }
    const float invS = 1.0f / S;
    const float pp   = T2 * invS * invS;       // sum probs^2
    const float ph   = T1 * invS;              // sum probs*hist
    const float dot  = seen[id] ? fmaf(EMA_OLD, ph, EMA_NEW * pp) : pp;
    const float elr  = __logf(1.0f - dot + EPS_ELR);
    const float ce   = m + __logf(S) - s_logits[targets[row]];
    partial[row] = ce + LAMBDA_ELR * elr;
  }
}

// Deterministic single-block tree reduction of the per-row partials.
__global__ __launch_bounds__(1024)
void elr_final_reduce(const float* __restrict__ partial,
                      float* __restrict__ out, int B)
{
  __shared__ float s_red[32];
  const int tid = threadIdx.x;
  float acc = 0.f;
  for (int i = tid; i < B; i += 1024) acc += partial[i];
  acc = wave_sum(acc);
  if ((tid & 31) == 0) s_red[tid >> 5] = acc;
  __syncthreads();
  if (tid == 0) {
    float t = 0.f;
#pragma unroll
    for (int w = 0; w < 32; ++w) t += s_red[w];
    out[0] = t / (float)B;
  }
}

extern "C" void kernel_launch(void* const* d_in, const int* in_sizes, int n_in,
                              void* d_out, int out_size, void* d_ws, size_t ws_size,
                              hipStream_t stream) {
  // setup_inputs() order: logits[B,C] f32, history[N,C] f32, seen[N] bool,
  // targets[B] i32, ids[B] int.
  const float*         logits  = (const float*)d_in[0];
  const float*         history = (const float*)d_in[1];
  const unsigned char* seen    = (const unsigned char*)d_in[2];
  const int*           targets = (const int*)d_in[3];
  const int*           ids     = (const int*)d_in[4];

  const int B = in_sizes[3];            // targets has shape [B]
  const int C = in_sizes[0] / B;        // logits is [B, C]

  float* partial = (float*)d_ws;        // B floats of scratch

  elr_row_kernel<<<B, BLOCK_THREADS, 0, stream>>>(
      logits, history, seen, targets, ids, partial, C);
  elr_final_reduce<<<1, 1024, 0, stream>>>(partial, (float*)d_out, B);
}